// HSR_DIGNET_35545149341818
// MI455X (gfx1250) — compile-verified
//
#include <hip/hip_runtime.h>

#define N_NODES 50000
#define N_EDGES 800000
#define D 96
#define LDA 100   // padded LDS row stride (DWORDs): 100 mod 64 = 36, gcd(36,64)=4 -> 16 distinct banks

typedef __attribute__((ext_vector_type(2))) float v2f;
typedef __attribute__((ext_vector_type(8))) float v8f;

// ---------------- Kernel 1: zero the aggregation buffer ----------------
__global__ void cgc_zero_kernel(float4* __restrict__ aggr) {
    int i = blockIdx.x * 256 + threadIdx.x;
    if (i < (N_NODES * D) / 4) {
        aggr[i] = make_float4(0.f, 0.f, 0.f, 0.f);
    }
}

// ---------------- Kernel 2: edge gather-multiply + scatter-add ----------------
// blockDim = (96, 4): threadIdx.x = feature (3 waves, coalesced 128B per wave),
// threadIdx.y = edge-in-block. Per-edge scalars are wave-uniform.
__global__ void cgc_scatter_kernel(const float* __restrict__ x,
                                   const int*   __restrict__ ei,
                                   const float* __restrict__ ew,
                                   const float* __restrict__ pr,
                                   float*       __restrict__ aggr) {
    int e = blockIdx.x * 4 + threadIdx.y;
    if (e >= N_EDGES) return;
    int f = threadIdx.x;
    int src = ei[e];              // row 0 of edge_index
    int dst = ei[N_EDGES + e];    // row 1 of edge_index
    float w = ew[e] * pr[src];    // wave-uniform scalar
    float m = w * x[src * D + f];
    // hardware global_atomic_add_f32 (no return) -> L2 atomic units
    unsafeAtomicAdd(&aggr[dst * D + f], m);
}

// ---------------- Kernel 3: out = (aggr + x) @ W^T + b via V_WMMA_F32_16X16X4_F32 ----
// One block = one 16-node tile. 6 waves (192 threads) = 6 output-column tiles (96 = 6*16).
// A tile (aggr+x, 16x96) staged once in LDS, reused by all 6 waves.
// B operand = W^T tile: b.x/b.y are contiguous W[n][k], W[n][k+1] -> float2 global loads.
__global__ void __launch_bounds__(192)
cgc_gemm_kernel(const float* __restrict__ aggr,
                const float* __restrict__ x,
                const float* __restrict__ W,
                const float* __restrict__ bias,
                float*       __restrict__ out) {
    __shared__ float As[16 * LDA];

    const int tile_m = blockIdx.x;        // 0..3124 (50000/16 exactly)
    const int tid    = threadIdx.x;       // 0..191
    const int base   = tile_m * 16 * D;

    // Cooperative LDS fill: A = aggr + x for 16 rows x 96 cols
    for (int i = tid; i < 16 * D; i += 192) {
        int r = i / D;
        int c = i - r * D;
        As[r * LDA + c] = aggr[base + i] + x[base + i];
    }
    __syncthreads();

    const int wave = tid >> 5;            // 0..5  == output column tile
    const int lane = tid & 31;
    const int half = lane >> 4;           // 0: K=0,1 ; 1: K=2,3 (f32 A/B operand layout)
    const int l16  = lane & 15;

    const int   col  = wave * 16 + l16;   // output column N for this lane
    const float* Wrow = W + col * D;      // W row-major: W[col][k]

    v8f c = {};                            // 16x16 f32 accumulator (8 VGPRs)
    const float* Arow = As + l16 * LDA;    // A row M = l16

    #pragma unroll
    for (int k = 0; k < D; k += 4) {
        v2f a;
        a.x = Arow[k + 2 * half + 0];      // A[M=l16][K=k+2*half]
        a.y = Arow[k + 2 * half + 1];      // A[M=l16][K=k+2*half+1]
        v2f bv;
        bv.x = Wrow[k + 2 * half + 0];     // B[K][N] = W[N][K], contiguous pair
        bv.y = Wrow[k + 2 * half + 1];
        // D = A(16x4) x B(4x16) + C   -> v_wmma_f32_16x16x4_f32
        c = __builtin_amdgcn_wmma_f32_16x16x4_f32(
                /*neg_a=*/false, a, /*neg_b=*/false, bv,
                /*c_mod=*/(short)0, c, /*reuse_a=*/false, /*reuse_b=*/false);
    }

    const float bv = bias[col];
    const int row0 = tile_m * 16 + half * 8;   // C/D layout: VGPR r -> M = r + 8*half
    #pragma unroll
    for (int r = 0; r < 8; ++r) {
        out[(row0 + r) * D + col] = c[r] + bv;
    }
}

extern "C" void kernel_launch(void* const* d_in, const int* in_sizes, int n_in,
                              void* d_out, int out_size, void* d_ws, size_t ws_size,
                              hipStream_t stream) {
    const float* x    = (const float*)d_in[0];
    const int*   ei   = (const int*)  d_in[1];
    const float* ew   = (const float*)d_in[2];
    const float* pr   = (const float*)d_in[3];
    const float* W    = (const float*)d_in[4];
    const float* bias = (const float*)d_in[5];
    float* out  = (float*)d_out;
    float* aggr = (float*)d_ws;   // N_NODES * D floats = 19.2 MB scratch

    // 1) aggr = 0
    cgc_zero_kernel<<<((N_NODES * D) / 4 + 255) / 256, 256, 0, stream>>>((float4*)aggr);

    // 2) scatter-add messages (4 edges per 384-thread block)
    dim3 sblk(96, 4);
    cgc_scatter_kernel<<<N_EDGES / 4, sblk, 0, stream>>>(x, ei, ew, pr, aggr);

    // 3) fused residual-add + GEMM + bias via f32 WMMA
    cgc_gemm_kernel<<<N_NODES / 16, 192, 0, stream>>>(aggr, x, W, bias, out);
}